// NeuSSampler_67551245631717
// MI455X (gfx1250) — compile-verified
//
#include <hip/hip_runtime.h>
#include <cstdint>
#include <cstddef>

// NeuS PDF importance sampler for MI455X (gfx1250).
// Memory-bound: ~170 MB traffic @ 23.3 TB/s => ~7.3 us floor. No matrix ops
// (WMMA n/a; fp32 scan precision required). CDNA5 paths used: wave32 shuffle
// scan, GLOBAL_LOAD_ASYNC_TO_LDS (ASYNCcnt) for bin edges, branchless LDS
// bisection, non-temporal streaming loads/stores.

#define HPAD 1e-5f   // PDFSampler histogram_padding
#define WEPS 1e-5f   // eps for weight-sum padding

typedef float vfloat4 __attribute__((ext_vector_type(4)));  // native vector

#if __has_builtin(__builtin_amdgcn_global_load_async_to_lds_b32)
#define HAVE_ASYNC_LDS 1
#else
#define HAVE_ASYNC_LDS 0
#endif

// Builtin signature (probe-learned): (addrspace(1) int*, addrspace(3) int*,
// imm offset, imm cpol).
typedef __attribute__((address_space(1))) int* gas1_int_ptr;
typedef __attribute__((address_space(3))) int* las3_int_ptr;

__device__ __forceinline__ void async_copy_f32(const float* g, float* l) {
#if HAVE_ASYNC_LDS
  __builtin_amdgcn_global_load_async_to_lds_b32(
      (gas1_int_ptr)(float*)g, (las3_int_ptr)l, /*imm offset*/ 0, /*cpol*/ 0);
#else
  *l = *g;  // fallback: synchronous copy through VGPRs
#endif
}

__device__ __forceinline__ void async_wait_all() {
#if HAVE_ASYNC_LDS
#if __has_builtin(__builtin_amdgcn_s_wait_asynccnt)
  __builtin_amdgcn_s_wait_asynccnt(0);
#else
  asm volatile("s_wait_asynccnt 0" ::: "memory");
#endif
  asm volatile("" ::: "memory");
#endif
}

// One wave (32 lanes) per ray. IPL = weights per lane = S/32.
template <int IPL>
__global__ __launch_bounds__(256) void neus_pdf_sample_kernel(
    const float* __restrict__ weights,  // [R, S]
    const float* __restrict__ ebins,    // [R, S+1]
    const float* __restrict__ nears,    // [R]
    const float* __restrict__ fars,     // [R]
    float* __restrict__ out,            // [R, nb]
    int R, int nb) {
  constexpr int S = IPL * 32;
  extern __shared__ float smem[];
  const int lane = threadIdx.x & 31;
  const int wid  = threadIdx.x >> 5;
  const int ray  = blockIdx.x * (blockDim.x >> 5) + wid;

  constexpr int STRIDE = S + 4;  // pad; keeps per-wave regions 16B aligned
  float* cdf_s = smem + (size_t)wid * 2 * STRIDE;
  float* bin_s = cdf_s + STRIDE;

  if (ray >= R) return;  // whole-wave exit; no cross-wave barriers used

  // ---- Stage bin edges into LDS asynchronously (overlaps with scan) ----
  const float* gb = ebins + (size_t)ray * (S + 1);
#pragma unroll
  for (int i = 0; i < IPL; ++i)
    async_copy_f32(gb + lane + 32 * i, bin_s + lane + 32 * i);
  if (lane == 0) async_copy_f32(gb + S, bin_s + S);  // edge S (the last one)

  // ---- Load weights (streaming/non-temporal; touch-once data) ----
  const float* gw = weights + (size_t)ray * S;
  float w[IPL];
  if constexpr (IPL == 4) {
    const vfloat4 v =
        __builtin_nontemporal_load((const vfloat4*)(gw + 4 * lane));
    w[0] = v.x; w[1] = v.y; w[2] = v.z; w[3] = v.w;
  } else {
#pragma unroll
    for (int j = 0; j < IPL; ++j)
      w[j] = __builtin_nontemporal_load(gw + lane * IPL + j);
  }

  // ---- Padded prefix sum: lane-local scan + wave32 shuffle scan ----
  float run = 0.f;
  float incl[IPL];
#pragma unroll
  for (int j = 0; j < IPL; ++j) { run += w[j] + HPAD; incl[j] = run; }
  const float laneSum = run;

  float scan = laneSum;
#pragma unroll
  for (int off = 1; off < 32; off <<= 1) {
    const float t = __shfl_up(scan, off, 32);
    scan += (lane >= off) ? t : 0.0f;
  }
  const float total = __shfl(scan, 31, 32);     // sum of (w + HPAD)
  const float excl  = scan - laneSum;           // exclusive lane prefix

  const float padding = fmaxf(0.f, WEPS - total);
  const float inv     = 1.f / (total + padding);
  const float padc    = padding * (1.f / (float)S);

  // cdf[0]=0; cdf[i+1] = min(1, (prefix_i + (i+1)*pad/S) / (total+pad))
  if (lane == 0) cdf_s[0] = 0.f;
#pragma unroll
  for (int j = 0; j < IPL; ++j) {
    const int gi = lane * IPL + j;
    const float c = (excl + incl[j] + (float)(gi + 1) * padc) * inv;
    cdf_s[gi + 1] = fminf(1.f, c);
  }

  const float nearv = nears[ray];
  const float farv  = fars[ray];
  const float span  = farv - nearv;
  // u_q = q * (1 - 1/nb)/(nb-1) + 1/(2nb)   (reference linspace + half-bin)
  const float step = (nb > 1) ? (1.0f - 1.0f / (float)nb) / (float)(nb - 1) : 0.f;
  const float half = 0.5f / (float)nb;

  async_wait_all();  // bin edges resident in LDS from here on

  // ---- Inverse CDF: branchless unrolled bisection over LDS cdf ----
  float* orow = out + (size_t)ray * nb;
  for (int q = lane; q < nb; q += 32) {
    const float u = fmaf((float)q, step, half);
    // Find last index lo in [0, S] with cdf[lo] <= u (cdf[-1] = -inf).
    // searchsorted(side='right') result is then inds = lo + 1.
    int lo = -1;
#pragma unroll
    for (int st = S; st > 0; st >>= 1) {       // S, S/2, ..., 1 (log2(S)+1 steps)
      const int cand = lo + st;
      const bool adv = (cand <= S) && (cdf_s[cand] <= u);
      lo = adv ? cand : lo;
    }
    const int inds = lo + 1;                    // in [0, S+1]
    int below = inds - 1; if (below < 0) below = 0; if (below > S) below = S;
    int above = inds;     if (above > S) above = S;

    const float c0 = cdf_s[below], c1 = cdf_s[above];
    const float b0 = bin_s[below], b1 = bin_s[above];
    const float d  = c1 - c0;
    float t = (d > 0.f) ? (u - c0) / d : 0.f;  // nan_to_num(nan/inf -> 0)
    t = fminf(fmaxf(t, 0.f), 1.f);             // clip(0,1)
    const float b = fmaf(t, b1 - b0, b0);
    __builtin_nontemporal_store(fmaf(b, span, nearv), orow + q);
  }
}

extern "C" void kernel_launch(void* const* d_in, const int* in_sizes, int n_in,
                              void* d_out, int out_size, void* d_ws, size_t ws_size,
                              hipStream_t stream) {
  (void)n_in; (void)d_ws; (void)ws_size;
  const float* weights = (const float*)d_in[0];
  const float* ebins   = (const float*)d_in[1];
  const float* nears   = (const float*)d_in[2];
  const float* fars    = (const float*)d_in[3];
  float* out = (float*)d_out;

  const int R  = in_sizes[2];                    // rays (nears count)
  const int S  = (R > 0) ? in_sizes[0] / R : 0;  // existing samples per ray
  const int nb = (R > 0) ? out_size / R : 0;     // num_samples + 1
  if (R <= 0 || S <= 0 || nb <= 0) return;

  constexpr int RPB = 8;  // rays (waves) per 256-thread block
  dim3 block(RPB * 32);
  dim3 grid((R + RPB - 1) / RPB);
  const size_t shmem = (size_t)RPB * 2 * (size_t)(S + 4) * sizeof(float);

  switch (S / 32) {
    case 1:
      hipLaunchKernelGGL((neus_pdf_sample_kernel<1>), grid, block, shmem, stream,
                         weights, ebins, nears, fars, out, R, nb);
      break;
    case 2:
      hipLaunchKernelGGL((neus_pdf_sample_kernel<2>), grid, block, shmem, stream,
                         weights, ebins, nears, fars, out, R, nb);
      break;
    case 8:
      hipLaunchKernelGGL((neus_pdf_sample_kernel<8>), grid, block, shmem, stream,
                         weights, ebins, nears, fars, out, R, nb);
      break;
    case 4:
    default:
      hipLaunchKernelGGL((neus_pdf_sample_kernel<4>), grid, block, shmem, stream,
                         weights, ebins, nears, fars, out, R, nb);
      break;
  }
}